// GreensFunctionLayer_74663711474146
// MI455X (gfx1250) — compile-verified
//
#include <hip/hip_runtime.h>
#include <math.h>

// ---------------------------------------------------------------------------
// GreensFunctionLayer on MI455X (gfx1250, wave32)
//
//   1) omega MLP -> w[b]                               (tiny, VALU)
//   2) A = (w_b + i*eta) I - H_sym                     (Re in d_ws, Im in d_out)
//   3) 32 batched complex 1024x1024 inversions, blocked in-place Gauss-Jordan
//      with 64-wide panels (16 steps). All O(n^3) work on
//      v_wmma_f32_16x16x4_f32 (full f32: the eta=0.01 resolvent is
//      ill-conditioned, bf16 would destroy the inverse).
//      - Panel width 64 keeps the dominant HBM cost (trailing C-plane
//        read+write sweeps) at ~8 GB (~0.35 ms @ 23.3 TB/s).
//      - Trailing waves compute a 64x16 C column (4 subtiles) sharing the
//        B operand: 0.75 VMEM issues per WMMA, so the loop is matrix-core-
//        bound rather than VMEM-issue-bound.
//   4) |G| elementwise -> d_out
// ---------------------------------------------------------------------------

typedef __attribute__((ext_vector_type(2))) float v2f;
typedef __attribute__((ext_vector_type(8))) float v8f;

#define NN     1024
#define NB     64            // panel width
#define NBLK   (NN / NB)     // 16 panel steps
#define NSUB   (NB / 16)     // 4 16-wide sub-tiles per panel
#define NCH    (NB / 4)      // 16 K-chunks of 4 per 64-deep dot
#define BATCH  32
#define ETA_F  0.01f

// ---------------- WMMA helpers (f32 16x16x4, wave32) -----------------------

__device__ __forceinline__ v8f wmma4(v2f a, v2f b, v8f c) {
  // (neg_a, A, neg_b, B, c_mod, C, reuse_a, reuse_b); f32 WMMA has no A/B NEG
  // (ISA: NEG = CNeg only), so operand signs are VALU-negated before the call.
  return __builtin_amdgcn_wmma_f32_16x16x4_f32(false, a, false, b, (short)0, c,
                                               false, false);
}

// A-operand 16x4 chunk of a 16xK tile at (row0,col0): lane L: M = L&15,
// K = 4c + 2*(L>=16) + {0,1}  (contiguous pair -> one b64 load)
__device__ __forceinline__ v2f loadA16(const float* base, int row0, int col0,
                                       int c, int l15, int half) {
  const float* p = base + (size_t)(row0 + l15) * NN + col0 + 4 * c + 2 * half;
  v2f r;
  r.x = p[0];
  r.y = p[1];
  return r;
}

// B-operand 4x16 chunk: lane L: N = L&15, K = 4c + 2*(L>=16) + {0,1}
__device__ __forceinline__ v2f loadB16(const float* base, int row0, int col0,
                                       int c, int l15, int half) {
  const float* p = base + (size_t)(row0 + 4 * c + 2 * half) * NN + col0 + l15;
  v2f r;
  r.x = p[0];
  r.y = p[NN];
  return r;
}

// C/D 16x16 tile: VGPR r holds row (r + 8*(L>=16)), column L&15
__device__ __forceinline__ v8f loadC16(const float* base, int row0, int col0,
                                       int l15, int half) {
  v8f c;
#pragma unroll
  for (int r = 0; r < 8; ++r)
    c[r] = base[(size_t)(row0 + r + 8 * half) * NN + col0 + l15];
  return c;
}

__device__ __forceinline__ void storeC16(float* base, int row0, int col0,
                                         int l15, int half, v8f c) {
#pragma unroll
  for (int r = 0; r < 8; ++r)
    base[(size_t)(row0 + r + 8 * half) * NN + col0 + l15] = c[r];
}

// ---------------- 1) omega MLP + mean ---------------------------------------

__global__ void __launch_bounds__(256)
omega_kernel(const float* __restrict__ gs, const float* __restrict__ W1,
             const float* __restrict__ b1, const float* __restrict__ W2,
             const float* __restrict__ b2, float* __restrict__ omega) {
  int t = blockIdx.x * blockDim.x + threadIdx.x;
  if (t >= BATCH * NN) return;
  const float* x = gs + (size_t)t * 64;
  float out = b2[0];
#pragma unroll 4
  for (int k = 0; k < 32; ++k) {
    float acc = b1[k];
#pragma unroll 8
    for (int d = 0; d < 64; ++d) acc += x[d] * W1[d * 32 + k];
    float s = acc / (1.0f + __expf(-acc));  // SiLU
    out += s * W2[k];
  }
  omega[t] = out;
}

__global__ void __launch_bounds__(256)
wmean_kernel(const float* __restrict__ omega, float* __restrict__ w) {
  __shared__ float sm[256];
  int b = blockIdx.x;
  float s = 0.0f;
  for (int g = threadIdx.x; g < NN; g += 256) s += omega[b * NN + g];
  sm[threadIdx.x] = s;
  __syncthreads();
  for (int off = 128; off > 0; off >>= 1) {
    if (threadIdx.x < off) sm[threadIdx.x] += sm[threadIdx.x + off];
    __syncthreads();
  }
  if (threadIdx.x == 0) w[b] = sm[0] * (1.0f / (float)NN);
}

// ---------------- 2) build resolvent ----------------------------------------

__global__ void __launch_bounds__(256)
init_kernel(const float* __restrict__ H, const float* __restrict__ w,
            float* __restrict__ Ar, float* __restrict__ Ai) {
  size_t idx = (size_t)blockIdx.x * blockDim.x + threadIdx.x;
  if (idx >= (size_t)BATCH * NN * NN) return;
  int b = (int)(idx >> 20);
  int rem = (int)(idx & (NN * NN - 1));
  int i = rem >> 10, j = rem & (NN - 1);
  float hs = 0.5f * (H[(size_t)i * NN + j] + H[(size_t)j * NN + i]);
  Ar[idx] = (i == j ? w[b] : 0.0f) - hs;
  Ai[idx] = (i == j ? ETA_F : 0.0f);
}

// ---------------- 3) blocked Gauss-Jordan, NB = 64 --------------------------

// 64x64 complex pivot inverse in LDS (unblocked GJ, no pivoting; the i*eta
// shift keeps it nonsingular). One workgroup/batch; 256 threads own 16
// elements each. 4-phase barrier discipline per pivot keeps the in-place
// ordering (new row-p, old col-p) race-free. Rows padded to 65 floats so
// column reads hit 64 distinct LDS banks.
__global__ void __launch_bounds__(256)
gj_pivot(float* __restrict__ Ar, float* __restrict__ Ai, int kb) {
  __shared__ float sr[NB][NB + 1];
  __shared__ float si[NB][NB + 1];
  int b = blockIdx.x;
  int t = threadIdx.x;
  float* ar = Ar + (size_t)b * NN * NN;
  float* ai = Ai + (size_t)b * NN * NN;
  int c0 = kb * NB;

#pragma unroll
  for (int s = 0; s < 16; ++s) {
    int e = t + 256 * s;
    int i = e >> 6, j = e & 63;
    size_t g = (size_t)(c0 + i) * NN + c0 + j;
    sr[i][j] = ar[g];
    si[i][j] = ai[g];
  }
  __syncthreads();

  for (int p = 0; p < NB; ++p) {
    float dr = sr[p][p], di = si[p][p];
    float inv = 1.0f / (dr * dr + di * di);
    float pvr = dr * inv, pvi = -di * inv;  // piv = 1/d
    __syncthreads();                        // all threads captured d
    // Phase A: scale row p (and write pivot)
    if (t < NB) {
      int j = t;
      if (j == p) {
        sr[p][p] = pvr;
        si[p][p] = pvi;
      } else {
        float r = sr[p][j], im = si[p][j];
        sr[p][j] = r * pvr - im * pvi;
        si[p][j] = r * pvi + im * pvr;
      }
    }
    __syncthreads();                        // row p now new
    // Phase B: a_ij -= a_ip(old) * a_pj(new)   for i!=p, j!=p
#pragma unroll
    for (int s = 0; s < 16; ++s) {
      int e = t + 256 * s;
      int i = e >> 6, j = e & 63;
      if (i != p && j != p) {
        float xr = sr[i][p], xi = si[i][p];
        float yr = sr[p][j], yi = si[p][j];
        sr[i][j] -= xr * yr - xi * yi;
        si[i][j] -= xr * yi + xi * yr;
      }
    }
    __syncthreads();                        // old column p fully consumed
    // Phase C: a_ip = -a_ip(old) * piv
    if (t < NB && t != p) {
      int i = t;
      float xr = sr[i][p], xi = si[i][p];
      sr[i][p] = -(xr * pvr - xi * pvi);
      si[i][p] = -(xr * pvi + xi * pvr);
    }
    __syncthreads();
  }

#pragma unroll
  for (int s = 0; s < 16; ++s) {
    int e = t + 256 * s;
    int i = e >> 6, j = e & 63;
    size_t g = (size_t)(c0 + i) * NN + c0 + j;
    ar[g] = sr[i][j];
    ai[g] = si[i][j];
  }
}

// row panel: A_kj <- P * A_kj. One wave OWNS an entire 64x16 column strip:
// preload it (B operand, 16 chunks x Re/Im) into registers before any store,
// then stream P (A operand) from global. In-place safe within the wave.
__global__ void __launch_bounds__(32)
gj_row(float* __restrict__ Ar, float* __restrict__ Ai, int kb) {
  int jt = blockIdx.x, b = blockIdx.y;
  if ((jt >> 2) == kb) return;  // column tile inside pivot block
  int lane = threadIdx.x & 31;
  int half = lane >> 4, l15 = lane & 15;
  float* ar = Ar + (size_t)b * NN * NN;
  float* ai = Ai + (size_t)b * NN * NN;
  int ck = kb * NB, cj = jt * 16;

  v2f brr[NCH], bii[NCH];
#pragma unroll
  for (int c = 0; c < NCH; ++c) {
    brr[c] = loadB16(ar, ck, cj, c, l15, half);
    bii[c] = loadB16(ai, ck, cj, c, l15, half);
  }

  v8f zr = {0.f, 0.f, 0.f, 0.f, 0.f, 0.f, 0.f, 0.f};
#pragma unroll
  for (int r = 0; r < NSUB; ++r) {
    int row0 = ck + r * 16;
    v8f accR = zr, accI = zr;
#pragma unroll
    for (int c = 0; c < NCH; ++c) {
      v2f prr = loadA16(ar, row0, ck, c, l15, half);  // P rows for this sub
      v2f pii = loadA16(ai, row0, ck, c, l15, half);
      v2f npii = -pii;
      // Re = Pr*Br - Pi*Bi ; Im = Pr*Bi + Pi*Br
      accR = wmma4(prr, brr[c], accR);
      accR = wmma4(npii, bii[c], accR);
      accI = wmma4(prr, bii[c], accI);
      accI = wmma4(pii, brr[c], accI);
    }
    storeC16(ar, row0, cj, l15, half, accR);
    storeC16(ai, row0, cj, l15, half, accI);
  }
}

// trailing update (the n^3 bulk): C -= A_ik * A_kj, K = 64.
// 8 waves/block; each wave computes a 64x16 C column (4 row-subtiles, quad-
// aligned so the pivot-row skip is uniform) sharing the B operand chunk:
// 12 VMEM issues per 16 WMMAs.
__global__ void __launch_bounds__(256)
gj_trailing(float* __restrict__ Ar, float* __restrict__ Ai, int kb) {
  int iq = blockIdx.y, b = blockIdx.z;
  if (iq == kb) return;  // whole row-quad inside pivot block
  int wave = threadIdx.x >> 5;
  int jt = blockIdx.x * 8 + wave;
  if ((jt >> 2) == kb) return;  // wave-uniform exit
  int lane = threadIdx.x & 31;
  int half = lane >> 4, l15 = lane & 15;
  float* ar = Ar + (size_t)b * NN * NN;
  float* ai = Ai + (size_t)b * NN * NN;
  int rq = iq * NB, ck = kb * NB, cj = jt * 16;

  v8f accR[NSUB], accI[NSUB];
#pragma unroll
  for (int s = 0; s < NSUB; ++s) {
    accR[s] = loadC16(ar, rq + s * 16, cj, l15, half);
    accI[s] = loadC16(ai, rq + s * 16, cj, l15, half);
  }

#pragma unroll
  for (int c = 0; c < NCH; ++c) {
    v2f brr = loadB16(ar, ck, cj, c, l15, half);  // A_kj (new row panel)
    v2f bii = loadB16(ai, ck, cj, c, l15, half);
#pragma unroll
    for (int s = 0; s < NSUB; ++s) {
      int r0 = rq + s * 16;
      v2f arr = loadA16(ar, r0, ck, c, l15, half);  // A_ik (old)
      v2f aii = loadA16(ai, r0, ck, c, l15, half);
      v2f narr = -arr, naii = -aii;
      // Re -= Ar*Br - Ai*Bi ; Im -= Ar*Bi + Ai*Br
      accR[s] = wmma4(narr, brr, accR[s]);
      accR[s] = wmma4(aii, bii, accR[s]);
      accI[s] = wmma4(narr, bii, accI[s]);
      accI[s] = wmma4(naii, brr, accI[s]);
    }
  }

#pragma unroll
  for (int s = 0; s < NSUB; ++s) {
    storeC16(ar, rq + s * 16, cj, l15, half, accR[s]);
    storeC16(ai, rq + s * 16, cj, l15, half, accI[s]);
  }
}

// column panel: A_ik <- -A_ik * P. One wave OWNS an entire 16x64 row strip:
// preload A_ik (A operand) into registers before any store, then stream P
// (B operand) and emit the 4 output sub-tiles.
__global__ void __launch_bounds__(32)
gj_col(float* __restrict__ Ar, float* __restrict__ Ai, int kb) {
  int it = blockIdx.x, b = blockIdx.y;
  if ((it >> 2) == kb) return;  // row strip inside pivot block
  int lane = threadIdx.x & 31;
  int half = lane >> 4, l15 = lane & 15;
  float* ar = Ar + (size_t)b * NN * NN;
  float* ai = Ai + (size_t)b * NN * NN;
  int r0 = it * 16, ck = kb * NB;

  v2f arr[NCH], aii[NCH];
#pragma unroll
  for (int c = 0; c < NCH; ++c) {
    arr[c] = loadA16(ar, r0, ck, c, l15, half);
    aii[c] = loadA16(ai, r0, ck, c, l15, half);
  }

  v8f zr = {0.f, 0.f, 0.f, 0.f, 0.f, 0.f, 0.f, 0.f};
#pragma unroll
  for (int c0 = 0; c0 < NSUB; ++c0) {
    int col0 = ck + c0 * 16;
    v8f accR = zr, accI = zr;
#pragma unroll
    for (int c = 0; c < NCH; ++c) {
      v2f prr = loadB16(ar, ck, col0, c, l15, half);  // P as B operand
      v2f pii = loadB16(ai, ck, col0, c, l15, half);
      v2f narr = -arr[c], naii = -aii[c];
      // Re = -(Ar*Pr - Ai*Pi) ; Im = -(Ar*Pi + Ai*Pr)
      accR = wmma4(narr, prr, accR);
      accR = wmma4(aii[c], pii, accR);
      accI = wmma4(narr, pii, accI);
      accI = wmma4(naii, prr, accI);
    }
    storeC16(ar, r0, col0, l15, half, accR);
    storeC16(ai, r0, col0, l15, half, accI);
  }
}

// ---------------- 4) |G| ----------------------------------------------------

__global__ void __launch_bounds__(256)
abs_kernel(const float* __restrict__ Ar, float* __restrict__ AiOut) {
  size_t idx = (size_t)blockIdx.x * blockDim.x + threadIdx.x;
  if (idx >= (size_t)BATCH * NN * NN) return;
  float r = Ar[idx], im = AiOut[idx];
  AiOut[idx] = sqrtf(r * r + im * im);
}

// ---------------- host launcher ---------------------------------------------

extern "C" void kernel_launch(void* const* d_in, const int* in_sizes, int n_in,
                              void* d_out, int out_size, void* d_ws,
                              size_t ws_size, hipStream_t stream) {
  const float* gs = (const float*)d_in[0];  // [32,1024,64]
  const float* H  = (const float*)d_in[1];  // [1024,1024]
  const float* W1 = (const float*)d_in[2];  // [64,32]
  const float* b1 = (const float*)d_in[3];  // [32]
  const float* W2 = (const float*)d_in[4];  // [32,1]
  const float* b2 = (const float*)d_in[5];  // [1]

  char* ws = (char*)d_ws;
  float* Ar    = (float*)ws;                                  // 128 MB: Re(A)
  float* omega = (float*)(ws + (size_t)BATCH * NN * NN * 4);  // 128 KB
  float* w     = omega + BATCH * NN;                          // 128 B
  float* Ai    = (float*)d_out;                               // Im(A) in d_out

  omega_kernel<<<(BATCH * NN + 255) / 256, 256, 0, stream>>>(gs, W1, b1, W2, b2,
                                                             omega);
  wmean_kernel<<<BATCH, 256, 0, stream>>>(omega, w);

  int nelem_blocks = (int)(((size_t)BATCH * NN * NN) / 256);
  init_kernel<<<nelem_blocks, 256, 0, stream>>>(H, w, Ar, Ai);

  for (int kb = 0; kb < NBLK; ++kb) {
    gj_pivot<<<BATCH, 256, 0, stream>>>(Ar, Ai, kb);
    gj_row<<<dim3(NN / 16, BATCH), 32, 0, stream>>>(Ar, Ai, kb);
    gj_trailing<<<dim3(8, NBLK, BATCH), 256, 0, stream>>>(Ar, Ai, kb);
    gj_col<<<dim3(NN / 16, BATCH), 32, 0, stream>>>(Ar, Ai, kb);
  }

  abs_kernel<<<nelem_blocks, 256, 0, stream>>>(Ar, Ai);
}